// Proximity_52896817218211
// MI455X (gfx1250) — compile-verified
//
#include <hip/hip_runtime.h>
#include <hip/hip_bf16.h>
#include <stdint.h>

typedef float v2f __attribute__((ext_vector_type(2)));
typedef float v8f __attribute__((ext_vector_type(8)));

static constexpr int   kB        = 16384;   // rows of x
static constexpr int   kD        = 1024;    // feature dim
static constexpr int   kBlkGrid  = 1024;    // blocks in partial kernel
static constexpr int   kBDim     = 256;     // 8 waves (wave32)
static constexpr int   kWaves    = 8;
static constexpr int   kRowHalf  = kBlkGrid * kWaves;  // 8192: row stride between the 2 rows per wave
static constexpr float kEps      = 1e-12f;
static constexpr float kClampMax = 1e12f;

// Generic pointers to LDS carry the wave-relative LDS byte offset in the low
// 32 bits (flat LDS aperture).
__device__ __forceinline__ uint32_t lds_byte_off(const void* p) {
  return (uint32_t)(uintptr_t)p;
}

// gfx1250 async copy global -> LDS (ASYNCcnt). GV mode: vdst = LDS byte
// offset VGPR, vaddr = 64-bit global address VGPR pair.
__device__ __forceinline__ void async_copy_b128(uint32_t lds_off, const void* gaddr) {
  asm volatile("global_load_async_to_lds_b128 %0, %1, off"
               :: "v"(lds_off), "v"((uint64_t)(uintptr_t)gaddr)
               : "memory");
}

// Threshold waits on the per-wave ASYNCcnt: async loads complete in order, so
// waiting to <=8 guarantees the first 8 issues (buffer 0) have landed while
// buffer 1's HBM traffic stays in flight.
__device__ __forceinline__ void wait_async_le8() {
  asm volatile("s_wait_asynccnt 0x8" ::: "memory");
}
__device__ __forceinline__ void wait_async_0() {
  asm volatile("s_wait_asynccnt 0x0" ::: "memory");
}

__device__ __forceinline__ float wave_sum32(float v) {
#pragma unroll
  for (int m = 16; m >= 1; m >>= 1) v += __shfl_xor(v, m, 32);
  return v;  // full xor-butterfly: every lane holds the total
}

// Per-row clamped squared distance between normalized vectors, given the
// wave's staged x row (LDS) and the pre-loaded center row (VGPRs).
__device__ __forceinline__ float row_dist(const float4* __restrict__ sx4,
                                          const float4* __restrict__ cv,
                                          int lane) {
  float sxx = 0.0f, scc = 0.0f, sxc = 0.0f;
#pragma unroll
  for (int it = 0; it < 8; ++it) {
    const float4 xv = sx4[it * 32 + lane];   // ds_load_b128
    const float4 c  = cv[it];
    sxx = fmaf(xv.x, xv.x, sxx); sxx = fmaf(xv.y, xv.y, sxx);
    sxx = fmaf(xv.z, xv.z, sxx); sxx = fmaf(xv.w, xv.w, sxx);
    scc = fmaf(c.x,  c.x,  scc); scc = fmaf(c.y,  c.y,  scc);
    scc = fmaf(c.z,  c.z,  scc); scc = fmaf(c.w,  c.w,  scc);
    sxc = fmaf(xv.x, c.x,  sxc); sxc = fmaf(xv.y, c.y,  sxc);
    sxc = fmaf(xv.z, c.z,  sxc); sxc = fmaf(xv.w, c.w,  sxc);
  }
  sxx = wave_sum32(sxx);
  scc = wave_sum32(scc);
  sxc = wave_sum32(sxc);
  const float nx = fmaxf(sqrtf(sxx), kEps);
  const float nc = fmaxf(sqrtf(scc), kEps);
  float dist = sxx / (nx * nx) + scc / (nc * nc) - 2.0f * (sxc / (nx * nc));
  return fminf(fmaxf(dist, kEps), kClampMax);
}

// ---------------------------------------------------------------------------
// Kernel 1: one wave per row, two rows per wave, software-pipelined through a
// 2-deep LDS double buffer fed by async global->LDS b128 copies. Center rows
// (4MB table, L2-resident) are fetched with regular global_load_b128 *before*
// any async wait so L2 traffic overlaps the HBM stream.
// ---------------------------------------------------------------------------
__global__ __launch_bounds__(kBDim) void prox_partial_kernel(
    const float* __restrict__ x,
    const int*   __restrict__ labels,
    const float* __restrict__ centers,
    float*       __restrict__ partial) {
  __shared__ __align__(16) float sx[2][kWaves][kD];   // exactly 64 KB

  const int tid  = threadIdx.x;
  const int wave = tid >> 5;
  const int lane = tid & 31;

  const int rowA = blockIdx.x * kWaves + wave;
  const int rowB = rowA + kRowHalf;

  const int labA = labels[rowA];
  const int labB = labels[rowB];
  const float* xrA = x + (size_t)rowA * kD;
  const float* xrB = x + (size_t)rowB * kD;
  const float4* crA4 = (const float4*)(centers + (size_t)labA * kD);
  const float4* crB4 = (const float4*)(centers + (size_t)labB * kD);

  const uint32_t sxA = lds_byte_off(&sx[0][wave][0]);
  const uint32_t sxB = lds_byte_off(&sx[1][wave][0]);

  // Issue all 16 async HBM->LDS copies first (8 per row buffer).
#pragma unroll
  for (int it = 0; it < 8; ++it) {
    const uint32_t off = (uint32_t)(it * 512 + lane * 16);
    async_copy_b128(sxA + off, (const char*)xrA + off);
  }
#pragma unroll
  for (int it = 0; it < 8; ++it) {
    const uint32_t off = (uint32_t)(it * 512 + lane * 16);
    async_copy_b128(sxB + off, (const char*)xrB + off);
  }

  // Overlap: pull both center rows (L2) while the async stream is in flight.
  float4 cA[8], cB[8];
#pragma unroll
  for (int it = 0; it < 8; ++it) cA[it] = crA4[it * 32 + lane];
#pragma unroll
  for (int it = 0; it < 8; ++it) cB[it] = crB4[it * 32 + lane];

  wait_async_le8();   // buffer 0 landed; buffer 1 still streaming
  const float dA = row_dist((const float4*)&sx[0][wave][0], cA, lane);

  wait_async_0();     // buffer 1 landed
  const float dB = row_dist((const float4*)&sx[1][wave][0], cB, lane);

  const float wsum = dA + dB;   // identical across the wave (full butterfly)

  // Block reduction: reuse the stage buffer as scratch (keeps LDS at 64 KB).
  __syncthreads();                       // all waves done reading sx
  float* wdist = &sx[0][0][0];
  if (lane == 0) wdist[wave] = wsum;
  __syncthreads();
  if (tid == 0) {
    float s = 0.0f;
#pragma unroll
    for (int w = 0; w < kWaves; ++w) s += wdist[w];
    partial[blockIdx.x] = s;
  }
}

// ---------------------------------------------------------------------------
// Kernel 2: reduce kBlkGrid partials with V_WMMA_F32_16X16X4_F32 (verified to
// lower last round). Each wmma folds a 16x4 chunk against a ones 4x16 B into
// the f32 accumulator; after all chunks, column 0 summed over M = grand total.
// Single full wave (EXEC all ones, as WMMA requires).
// ---------------------------------------------------------------------------
__global__ __launch_bounds__(32) void prox_reduce_kernel(
    const float* __restrict__ partial, float* __restrict__ out) {
  const int lane = threadIdx.x;        // 0..31, full wave
  const int m    = lane & 15;          // A-matrix row for this lane
  const int kb   = (lane >> 4) << 1;   // lanes 0-15 -> K{0,1}, lanes 16-31 -> K{2,3}

  v2f ones; ones[0] = 1.0f; ones[1] = 1.0f;
  v8f acc = {0.f, 0.f, 0.f, 0.f, 0.f, 0.f, 0.f, 0.f};

#pragma unroll
  for (int c = 0; c < kBlkGrid / 64; ++c) {
    const float* p = partial + c * 64;
    v2f a;
    a[0] = p[m * 4 + kb];
    a[1] = p[m * 4 + kb + 1];
    acc = __builtin_amdgcn_wmma_f32_16x16x4_f32(
        /*neg_a=*/false, a, /*neg_b=*/false, ones,
        /*c_mod=*/(short)0, acc, /*reuse_a=*/false, /*reuse_b=*/false);
  }

  // C/D layout: lane j (j<16), VGPR r -> D[M=r][N=j]; lane j+16 -> D[M=8+r][N=j].
  float s = acc[0] + acc[1] + acc[2] + acc[3] + acc[4] + acc[5] + acc[6] + acc[7];
  s += __shfl_xor(s, 16, 32);          // combine M=0..7 with M=8..15 halves
  if (lane == 0) out[0] = s * (1.0f / (float)kB);
}

extern "C" void kernel_launch(void* const* d_in, const int* in_sizes, int n_in,
                              void* d_out, int out_size, void* d_ws, size_t ws_size,
                              hipStream_t stream) {
  (void)in_sizes; (void)n_in; (void)out_size; (void)ws_size;
  const float* x       = (const float*)d_in[0];
  const int*   labels  = (const int*)d_in[1];
  const float* centers = (const float*)d_in[2];
  float*       partial = (float*)d_ws;             // kBlkGrid floats (4 KB)
  float*       out     = (float*)d_out;

  prox_partial_kernel<<<kBlkGrid, kBDim, 0, stream>>>(x, labels, centers, partial);
  prox_reduce_kernel<<<1, 32, 0, stream>>>(partial, out);
}